// SpectralConv2d_3307124818088
// MI455X (gfx1250) — compile-verified
//
#include <hip/hip_runtime.h>

// ---------------------------------------------------------------------------
// FNO SpectralConv2d for gfx1250 (MI455X): all stages are f32 WMMA GEMMs
// (V_WMMA_F32_16X16X4_F32). The channel-mix kernel streams the 268MB weight
// tensor via GLOBAL_LOAD_ASYNC_TO_LDS_B128 (ASYNCcnt) with double buffering,
// overlapping the dominant HBM stream with WMMA compute.
// ---------------------------------------------------------------------------

typedef float v2f __attribute__((ext_vector_type(2)));
typedef float v8f __attribute__((ext_vector_type(8)));

// twiddle table offsets (floats) inside workspace
constexpr int OFF_FWC  = 0;        // [128w][32k]  cos(2pi wk/128)
constexpr int OFF_FWMS = 4096;     // [128w][32k] -sin
constexpr int OFF_FHC  = 8192;     // [64r][128h]  cos(2pi row(r) h/128)
constexpr int OFF_FHS  = 16384;    // [64r][128h]  sin
constexpr int OFF_C2   = 24576;    // [128h][64r]  cos (inverse, e^{+i})
constexpr int OFF_S2   = 32768;    // [128h][64r]  sin
constexpr int OFF_CW2  = 40960;    // [32k][128w]  a_k cos /16384
constexpr int OFF_SW2M = 45056;    // [32k][128w] -a_k sin /16384
constexpr int TW_FLOATS = 49152;

__device__ __forceinline__ v8f wmma4(v2f a, v2f b, v8f c) {
  // D = A(16x4 f32) * B(4x16 f32) + C(16x16 f32)
  return __builtin_amdgcn_wmma_f32_16x16x4_f32(false, a, false, b,
                                               (short)0, c, false, false);
}

// A fragment (16x4): lane m=L&15 holds K = 2*(L>>4) .. +1
__device__ __forceinline__ v2f ldA(const float* A, int lda, int m0, int k0, int lane) {
  int m  = m0 + (lane & 15);
  int kk = k0 + ((lane >> 4) << 1);
  v2f r; r.x = A[m * lda + kk]; r.y = A[m * lda + kk + 1]; return r;
}
// B fragment (4x16): lane n=L&15 holds rows K = 2*(L>>4) .. +1
__device__ __forceinline__ v2f ldB(const float* B, int ldb, int k0, int n0, int lane) {
  int n  = n0 + (lane & 15);
  int kk = k0 + ((lane >> 4) << 1);
  v2f r; r.x = B[kk * ldb + n]; r.y = B[(kk + 1) * ldb + n]; return r;
}
// D store (16x16): VGPR j -> M = j + 8*(L>>4), N = L&15
__device__ __forceinline__ void stD(float* D, int ldd, int m0, int n0, int lane, v8f c) {
  int n  = n0 + (lane & 15);
  int mb = m0 + ((lane >> 4) << 3);
#pragma unroll
  for (int j = 0; j < 8; ++j) D[(mb + j) * ldd + n] = c[j];
}
__device__ __forceinline__ void stDneg(float* D, int ldd, int m0, int n0, int lane, v8f c) {
  int n  = n0 + (lane & 15);
  int mb = m0 + ((lane >> 4) << 3);
#pragma unroll
  for (int j = 0; j < 8; ++j) D[(mb + j) * ldd + n] = -c[j];
}

// Async memory->LDS copy of 16 bytes per lane (CDNA5 GLOBAL_LOAD_ASYNC_TO_LDS,
// tracked by ASYNCcnt). Generic->LDS pointer low 32 bits hold the LDS offset.
__device__ __forceinline__ void async_copy_b128(void* lds_dst, const void* gsrc) {
  unsigned int l = (unsigned int)(__UINTPTR_TYPE__)lds_dst;
  unsigned long long g = (unsigned long long)(__UINTPTR_TYPE__)gsrc;
  asm volatile("global_load_async_to_lds_b128 %0, %1, off"
               :: "v"(l), "v"(g) : "memory");
}
__device__ __forceinline__ void wait_async0() {
  asm volatile("s_wait_asynccnt 0x0" ::: "memory");
}

// ---------------------------------------------------------------------------
// Kernel 0: twiddle tables (tiny, one WG)
// ---------------------------------------------------------------------------
__global__ void fno_twiddle(float* __restrict__ tw) {
  const int tid = threadIdx.x;
  const float step = 6.28318530717958647692f / 128.0f;
  for (int idx = tid; idx < 4096; idx += 256) {
    int w = idx >> 5, k = idx & 31;
    float ang = step * (float)((w * k) & 127);
    tw[OFF_FWC + idx]  = cosf(ang);
    tw[OFF_FWMS + idx] = -sinf(ang);
  }
  for (int idx = tid; idx < 8192; idx += 256) {
    int r = idx >> 7, h = idx & 127;
    int row = (r < 32) ? r : (r + 64);          // rows {0..31, 96..127}
    float ang = step * (float)((row * h) & 127);
    tw[OFF_FHC + idx] = cosf(ang);
    tw[OFF_FHS + idx] = sinf(ang);
  }
  for (int idx = tid; idx < 8192; idx += 256) {
    int h = idx >> 6, r = idx & 63;
    int row = (r < 32) ? r : (r + 64);
    float ang = step * (float)((row * h) & 127);
    tw[OFF_C2 + idx] = cosf(ang);
    tw[OFF_S2 + idx] = sinf(ang);
  }
  for (int idx = tid; idx < 4096; idx += 256) {
    int k = idx >> 7, w = idx & 127;
    float s = ((k == 0) ? 1.0f : 2.0f) * (1.0f / 16384.0f); // fold 1/(H*W), rfft mirror
    float ang = step * (float)((k * w) & 127);
    tw[OFF_CW2 + idx]  = s * cosf(ang);
    tw[OFF_SW2M + idx] = -s * sinf(ang);
  }
}

// ---------------------------------------------------------------------------
// Kernel 1: forward partial DFT. One WG per (b,c).
//   T(128h x 32k) = x(128x128) @ FW  (complex via 2 real GEMMs)
//   X(64r  x 32k) = FH(64x128) @ T   (complex x complex)
// Output layout: Xre/Xim [b][c][r64][k32]
// LDS: xs[128][132] (reused for FH[2][64][66]) + Tre/Tim/Tneg [128][36]
// ---------------------------------------------------------------------------
__global__ void fno_fwd(const float* __restrict__ x, const float* __restrict__ tw,
                        float* __restrict__ Xre, float* __restrict__ Xim) {
  extern __shared__ float sm[];
  float* xs  = sm;                    // 128*132 = 16896
  float* Tre = sm + 16896;            // 128*36
  float* Tim = Tre + 4608;
  float* Trn = Tim + 4608;            // -T_re
  const int tid = threadIdx.x, lane = tid & 31, wv = tid >> 5;
  const int bc = blockIdx.x;
  const float* xg = x + (size_t)bc * 16384;

#pragma unroll
  for (int r = 0; r < 16; ++r) {      // 4096 float4s / 256 thr
    int j = r * 256 + tid;
    int row = j >> 5, c4 = (j & 31) << 2;
    *(float4*)(xs + row * 132 + c4) = *(const float4*)(xg + row * 128 + c4);
  }
  __syncthreads();

  { // Stage A: 16 tiles (8 Mt x 2 Nt), 2 per wave, K=128
    const float* FWc = tw + OFF_FWC;
    const float* FWm = tw + OFF_FWMS;
#pragma unroll
    for (int t = 0; t < 2; ++t) {
      int tile = t * 8 + wv;
      int m0 = (tile >> 1) << 4, n0 = (tile & 1) << 4;
      v8f ar = {}; v8f ai = {};
      for (int k0 = 0; k0 < 128; k0 += 4) {
        v2f a  = ldA(xs, 132, m0, k0, lane);
        v2f bc_ = ldB(FWc, 32, k0, n0, lane);
        v2f bm = ldB(FWm, 32, k0, n0, lane);
        ar = wmma4(a, bc_, ar);
        ai = wmma4(a, bm, ai);
      }
      stD(Tre, 36, m0, n0, lane, ar);
      stD(Tim, 36, m0, n0, lane, ai);
      stDneg(Trn, 36, m0, n0, lane, ar);
    }
  }
  __syncthreads();

  // reuse xs for FH tables (2 * 64*66 = 8448 <= 16896)
  float* FHc = xs; float* FHs = xs + 64 * 66;
  for (int idx = tid; idx < 8192; idx += 256) {
    int r = idx >> 7, h = idx & 127;
    FHc[r * 66 + h] = tw[OFF_FHC + idx];
    FHs[r * 66 + h] = tw[OFF_FHS + idx];
  }
  __syncthreads();

  { // Stage B: 8 tiles (4 Mt x 2 Nt), 1 per wave, K=128
    int m0 = (wv >> 1) << 4, n0 = (wv & 1) << 4;
    v8f ar = {}; v8f ai = {};
    for (int k0 = 0; k0 < 128; k0 += 4) {
      v2f ac = ldA(FHc, 66, m0, k0, lane);
      v2f as = ldA(FHs, 66, m0, k0, lane);
      v2f br = ldB(Tre, 36, k0, n0, lane);
      v2f bi = ldB(Tim, 36, k0, n0, lane);
      v2f bn = ldB(Trn, 36, k0, n0, lane);
      ar = wmma4(ac, br, ar);   // Xre = C@Tre + S@Tim
      ar = wmma4(as, bi, ar);
      ai = wmma4(ac, bi, ai);   // Xim = C@Tim - S@Tre
      ai = wmma4(as, bn, ai);
    }
    stD(Xre + (size_t)bc * 2048, 32, m0, n0, lane, ar);
    stD(Xim + (size_t)bc * 2048, 32, m0, n0, lane, ai);
  }
}

// ---------------------------------------------------------------------------
// Kernel 2: transpose Xre/Xim [b][i][mk2048] -> Xt [m][c][k][b][i]
// ---------------------------------------------------------------------------
__global__ void fno_t1(const float* __restrict__ Xre, const float* __restrict__ Xim,
                       float* __restrict__ Xt) {
  extern __shared__ float sm[];       // [32][65]
  const int b = blockIdx.x >> 2;
  const int i0 = (blockIdx.x & 3) << 5;
  const int tid = threadIdx.x;
  for (int c = 0; c < 2; ++c) {
    const float* src = c ? Xim : Xre;
    for (int mk0 = 0; mk0 < 2048; mk0 += 64) {
#pragma unroll
      for (int r = 0; r < 8; ++r) {
        int lin = r * 256 + tid;
        int ii = lin >> 6, u = lin & 63;
        sm[ii * 65 + u] = src[(size_t)(b * 128 + i0 + ii) * 2048 + mk0 + u];
      }
      __syncthreads();
#pragma unroll
      for (int r = 0; r < 8; ++r) {
        int lin = r * 256 + tid;
        int u = lin >> 5, ii = lin & 31;
        int mk = mk0 + u, m = mk >> 5, k = mk & 31;
        Xt[(size_t)(((m * 2 + c) * 32 + k) * 8 + b) * 128 + i0 + ii] = sm[ii * 65 + u];
      }
      __syncthreads();
    }
  }
}

// ---------------------------------------------------------------------------
// Kernel 3: channel mix. Grid (m=64, oq=4), 512 threads (16 waves).
// Per mode (m,k): A = [Xre(8b); Xim(8b)] (16x128i), B = Wre / Wim (128x128o).
// W streamed once in i-chunks of 4 with GLOBAL_LOAD_ASYNC_TO_LDS_B128 along
// its only contiguous axis ((k,c) runs of 256B), double-buffered so the HBM
// stream overlaps the WMMA compute. Output: Yre/Yim [m][k][b][o].
// LDS: W[2][4i][32o][68kc-pad] + X[2][32k][16row][6i-pad]
// ---------------------------------------------------------------------------
__global__ void fno_mix(const float* __restrict__ Xt,
                        const float* __restrict__ w1, const float* __restrict__ w2,
                        float* __restrict__ Yre, float* __restrict__ Yim) {
  extern __shared__ float sm[];
  float* const Wb0 = sm;              // 4*32*68 = 8704 each
  float* const Wb1 = sm + 8704;
  float* const Xb0 = sm + 17408;      // 32*16*6 = 3072 each
  float* const Xb1 = sm + 17408 + 3072;
  const int tid = threadIdx.x, lane = tid & 31, wv = tid >> 5;
  const int ot = wv & 1;              // o-tile within quarter
  const int kg = wv >> 1;             // k group (4 k each)
  const int m = blockIdx.x, oq = blockIdx.y;
  const float* Wg = (m < 32) ? w1 : w2;
  const int mx = (m < 32) ? m : (m - 32);
  const float* Xtm = Xt + (size_t)m * (2 * 32 * 8 * 128);

  // issue one W chunk (4 i-rows) as async copies: 2048 float4 / 512 thr
  auto issueW = [&](int i0, float* Wdst) {
#pragma unroll
    for (int r = 0; r < 4; ++r) {
      int lin = r * 512 + tid;
      int pr = lin >> 4, f4 = lin & 15;   // pr = il*32+ol
      int il = pr >> 5, ol = pr & 31;
      const float* g = Wg + (size_t)((i0 + il) * 128 + oq * 32 + ol) * 2048
                       + mx * 64 + f4 * 4;
      async_copy_b128(Wdst + pr * 68 + f4 * 4, g);
    }
  };
  // X chunk: [k32][row16][i4] (row = c*8+b), synchronous (tiny, L2-resident)
  auto loadX = [&](int i0, float* Xdst) {
#pragma unroll
    for (int r = 0; r < 4; ++r) {
      int lin = r * 512 + tid;
      int il = lin & 3, row = (lin >> 2) & 15, k = lin >> 6;
      Xdst[(k * 16 + row) * 6 + il] =
          Xtm[(size_t)(((row >> 3) * 32 + k) * 8 + (row & 7)) * 128 + i0 + il];
    }
  };

  v8f zero = {};
  v8f a1[4], a2[4];
#pragma unroll
  for (int j = 0; j < 4; ++j) { a1[j] = zero; a2[j] = zero; }

  issueW(0, Wb0);
  loadX(0, Xb0);
  wait_async0();
  __syncthreads();

  for (int i0 = 0; i0 < 128; i0 += 4) {
    const int cur = (i0 >> 2) & 1;
    float* const Wc = cur ? Wb1 : Wb0;
    float* const Xc = cur ? Xb1 : Xb0;
    if (i0 + 4 < 128) {               // prefetch next chunk into other buffer
      issueW(i0 + 4, cur ? Wb0 : Wb1);
      loadX(i0 + 4, cur ? Xb0 : Xb1);
    }
#pragma unroll
    for (int j = 0; j < 4; ++j) {     // WMMA on current chunk (overlaps async)
      int k = kg * 4 + j;
      v2f a = ldA(Xc + k * 96, 6, 0, 0, lane);
      int n = lane & 15, kk = (lane >> 4) << 1;
      int o = ot * 16 + n;
      int k2 = k * 2;
      const float* p = Wc + (kk * 32 + o) * 68;
      v2f br; br.x = p[k2];     br.y = p[2176 + k2];       // +32*68 = next i
      v2f bi; bi.x = p[k2 + 1]; bi.y = p[2176 + k2 + 1];
      a1[j] = wmma4(a, br, a1[j]);   // [Xre;Xim] @ Wre
      a2[j] = wmma4(a, bi, a2[j]);   // [Xre;Xim] @ Wim
    }
    wait_async0();                    // next chunk landed in LDS
    __syncthreads();
  }

  // combine halves: Yre = XreWre - XimWim ; Yim = XreWim + XimWre
#pragma unroll
  for (int j = 0; j < 4; ++j) {
    int k = kg * 4 + j;
#pragma unroll
    for (int jj = 0; jj < 8; ++jj) {
      float r1 = a1[j][jj], r2 = a2[j][jj];
      float r1x = __shfl_xor(r1, 16, 32);  // bring M=jj+8 (Xim rows) to lanes 0-15
      float r2x = __shfl_xor(r2, 16, 32);
      float yre = r1 - r2x;
      float yim = r2 + r1x;
      if (lane < 16) {
        size_t off = (size_t)((m * 32 + k) * 8 + jj) * 128 + oq * 32 + ot * 16 + lane;
        Yre[off] = yre;
        Yim[off] = yim;
      }
    }
  }
}

// ---------------------------------------------------------------------------
// Kernel 4: transpose Yre/Yim [mk][b][o] -> Yt [b][o][c][m][k]
// ---------------------------------------------------------------------------
__global__ void fno_t2(const float* __restrict__ Yre, const float* __restrict__ Yim,
                       float* __restrict__ Yt) {
  extern __shared__ float sm[];       // [32][65]
  const int b = blockIdx.x >> 2;
  const int o0 = (blockIdx.x & 3) << 5;
  const int tid = threadIdx.x;
  for (int c = 0; c < 2; ++c) {
    const float* src = c ? Yim : Yre;
    for (int mk0 = 0; mk0 < 2048; mk0 += 64) {
#pragma unroll
      for (int r = 0; r < 8; ++r) {
        int lin = r * 256 + tid;
        int u = lin >> 5, oo = lin & 31;
        sm[oo * 65 + u] = src[(size_t)(mk0 + u) * 1024 + b * 128 + o0 + oo];
      }
      __syncthreads();
#pragma unroll
      for (int r = 0; r < 8; ++r) {
        int lin = r * 256 + tid;
        int oo = lin >> 6, u = lin & 63;
        Yt[(size_t)(b * 128 + o0 + oo) * 4096 + c * 2048 + mk0 + u] = sm[oo * 65 + u];
      }
      __syncthreads();
    }
  }
}

// ---------------------------------------------------------------------------
// Kernel 5: inverse transform. One WG per (b,o).
//   G(128h x 32k)  = C2(128x64) @ Y  (complex, e^{+i} row ifft)
//   out(128x128)   = G_re @ CW2 + G_im @ SW2m  (irfft cols, scale folded)
// ---------------------------------------------------------------------------
__global__ void fno_inv(const float* __restrict__ Yt, const float* __restrict__ tw,
                        float* __restrict__ out) {
  extern __shared__ float sm[];
  float* Ysr = sm;                    // [64][36]
  float* Ysi = Ysr + 2304;
  float* Ysn = Ysi + 2304;            // -Yim
  float* G   = Ysn + 2304;            // [2][128][36]
  float* C2s = G + 9216;              // [128][68]
  float* S2s = C2s + 8704;
  const int tid = threadIdx.x, lane = tid & 31, wv = tid >> 5;
  const int bo = blockIdx.x;
  const float* Yg = Yt + (size_t)bo * 4096;

#pragma unroll
  for (int r = 0; r < 4; ++r) {       // 1024 float4 / 256 thr
    int f4i = r * 256 + tid;
    int c = f4i >> 9, rem = f4i & 511;
    int mrow = rem >> 3, k4 = (rem & 7) << 2;
    float4 v = *(const float4*)(Yg + f4i * 4);
    float* dst = (c ? Ysi : Ysr) + mrow * 36 + k4;
    *(float4*)dst = v;
    if (c) {
      float* d2 = Ysn + mrow * 36 + k4;
      d2[0] = -v.x; d2[1] = -v.y; d2[2] = -v.z; d2[3] = -v.w;
    }
  }
  for (int idx = tid; idx < 8192; idx += 256) {
    int h = idx >> 6, r = idx & 63;
    C2s[h * 68 + r] = tw[OFF_C2 + idx];
    S2s[h * 68 + r] = tw[OFF_S2 + idx];
  }
  __syncthreads();

  // Stage B': 32 jobs (c2 x 8Mt x 2Nt), 4 per wave, K=64
#pragma unroll
  for (int t = 0; t < 4; ++t) {
    int job = t * 8 + wv;
    int c = job >> 4, tile = job & 15;
    int m0 = (tile >> 1) << 4, n0 = (tile & 1) << 4;
    const float* B1 = c ? Ysi : Ysr;  // c0: Gre = C2@Yre + S2@(-Yim)
    const float* B2 = c ? Ysr : Ysn;  // c1: Gim = C2@Yim + S2@Yre
    v8f acc = {};
    for (int k0 = 0; k0 < 64; k0 += 4) {
      v2f ac = ldA(C2s, 68, m0, k0, lane);
      v2f as = ldA(S2s, 68, m0, k0, lane);
      v2f b1 = ldB(B1, 36, k0, n0, lane);
      v2f b2 = ldB(B2, 36, k0, n0, lane);
      acc = wmma4(ac, b1, acc);
      acc = wmma4(as, b2, acc);
    }
    stD(G + c * 4608, 36, m0, n0, lane, acc);
  }
  __syncthreads();

  // Stage A': 64 tiles (8x8), 8 per wave, K=32
  const float* CW = tw + OFF_CW2;
  const float* SW = tw + OFF_SW2M;
  float* og = out + (size_t)bo * 16384;
#pragma unroll
  for (int t = 0; t < 8; ++t) {
    int tile = t * 8 + wv;
    int m0 = (tile >> 3) << 4, n0 = (tile & 7) << 4;
    v8f acc = {};
    for (int k0 = 0; k0 < 32; k0 += 4) {
      v2f ar = ldA(G, 36, m0, k0, lane);
      v2f ai = ldA(G + 4608, 36, m0, k0, lane);
      v2f bc_ = ldB(CW, 128, k0, n0, lane);
      v2f bs = ldB(SW, 128, k0, n0, lane);
      acc = wmma4(ar, bc_, acc);
      acc = wmma4(ai, bs, acc);
    }
    stD(og, 128, m0, n0, lane, acc);
  }
}

// ---------------------------------------------------------------------------
extern "C" void kernel_launch(void* const* d_in, const int* in_sizes, int n_in,
                              void* d_out, int out_size, void* d_ws, size_t ws_size,
                              hipStream_t stream) {
  const float* x  = (const float*)d_in[0];   // [8][128][128][128]
  const float* w1 = (const float*)d_in[1];   // [128][128][32][32][2]
  const float* w2 = (const float*)d_in[2];
  float* out = (float*)d_out;                // [8][128][128][128]
  float* ws  = (float*)d_ws;                 // needs ~67 MB
  (void)in_sizes; (void)n_in; (void)out_size; (void)ws_size;

  float* tw  = ws;                           // 49152 floats of twiddles
  float* Xre = ws + TW_FLOATS;               // [b][i][64][32]   2,097,152
  float* Xim = Xre + 2097152;
  float* Xt  = Xim + 2097152;                // [m][c][k][b][i]  4,194,304
  float* Yre = Xt + 4194304;                 // [m][k][b][o]     2,097,152
  float* Yim = Yre + 2097152;
  float* Yt  = Yim + 2097152;                // [b][o][c][m][k]  4,194,304

  fno_twiddle<<<1, 256, 0, stream>>>(tw);
  fno_fwd<<<1024, 256, (16896 + 3 * 4608) * sizeof(float), stream>>>(x, tw, Xre, Xim);
  fno_t1<<<32, 256, 32 * 65 * sizeof(float), stream>>>(Xre, Xim, Xt);
  fno_mix<<<dim3(64, 4), 512, (2 * 8704 + 2 * 3072) * sizeof(float), stream>>>(
      Xt, w1, w2, Yre, Yim);
  fno_t2<<<32, 256, 32 * 65 * sizeof(float), stream>>>(Yre, Yim, Yt);
  fno_inv<<<1024, 256, (3 * 2304 + 9216 + 2 * 8704) * sizeof(float), stream>>>(Yt, tw, out);
}